// PrototypeLayer_2903397892433
// MI455X (gfx1250) — compile-verified
//
#include <hip/hip_runtime.h>
#include <hip/hip_bf16.h>
#include <math.h>

// Problem constants (from reference): B=1024, D=256, d=16, P=500
#define SZ_B 1024
#define SZ_D 256
#define SZ_d 16
#define SZ_P 500
#define NSWEEPS 6

typedef __attribute__((ext_vector_type(2))) float v2f;
typedef __attribute__((ext_vector_type(8))) float v8f;

// One wave computes M = X_b^T W_p for 4 prototypes (A cached in VGPRs),
// then runs one-sided Jacobi on two 16x16 matrices simultaneously
// (one per 16-lane half) to get singular values, sorts them descending
// (bitonic), and emits sum_k rel[k]*(1 - s_k).
__global__ __launch_bounds__(256)
void proto_grassmann_kernel(const float* __restrict__ xs,   // [B, D, d]
                            const float* __restrict__ xp,   // [P, D, d]
                            const float* __restrict__ rel,  // [1, d]
                            float* __restrict__ out)        // [B, P]
{
    const int lane   = threadIdx.x & 31;
    const int wave   = threadIdx.x >> 5;
    const int gw     = blockIdx.x * 8 + wave;      // global wave id
    const int b      = gw / (SZ_P / 4);            // 125 p-tiles per b
    const int p0     = (gw % (SZ_P / 4)) * 4;

    const int laneLo = lane & 15;
    const int laneHi = lane >> 4;                  // 0 or 1

    // ---- Phase 1a: cache A = X_b^T in registers (16x4 f32 A-layout) ----
    // A[m][K] with m = laneLo, K = 2*laneHi + {0,1} per 4-row chunk.
    const float* Xb = xs + (size_t)b * (SZ_D * SZ_d);
    v2f Areg[64];
    #pragma unroll
    for (int k = 0; k < 64; ++k) {
        int base = (4 * k + 2 * laneHi) * SZ_d + laneLo;
        Areg[k].x = Xb[base];
        Areg[k].y = Xb[base + SZ_d];
    }
    const float relk = rel[laneLo];

    for (int pp = 0; pp < 2; ++pp) {
        // ---- Phase 1b: two Gram matrices via chained f32 WMMA ----
        v8f M0 = {};
        v8f M1 = {};
        {
            const float* Wp = xp + (size_t)(p0 + pp * 2 + 0) * (SZ_D * SZ_d);
            #pragma unroll
            for (int k = 0; k < 64; ++k) {
                int base = (4 * k + 2 * laneHi) * SZ_d + laneLo;
                v2f bm; bm.x = Wp[base]; bm.y = Wp[base + SZ_d];
                M0 = __builtin_amdgcn_wmma_f32_16x16x4_f32(
                        false, Areg[k], false, bm, (short)0, M0, false, false);
            }
        }
        {
            const float* Wp = xp + (size_t)(p0 + pp * 2 + 1) * (SZ_D * SZ_d);
            #pragma unroll
            for (int k = 0; k < 64; ++k) {
                int base = (4 * k + 2 * laneHi) * SZ_d + laneLo;
                v2f bm; bm.x = Wp[base]; bm.y = Wp[base + SZ_d];
                M1 = __builtin_amdgcn_wmma_f32_16x16x4_f32(
                        false, Areg[k], false, bm, (short)0, M1, false, false);
            }
        }

        // ---- Redistribute: lane l<16 owns column l of M0; lane l>=16 owns
        // column (l-16) of M1. C-layout: VGPR j = row j (lanes 0-15) /
        // row j+8 (lanes 16-31), col = lane&15.
        float col[16];
        #pragma unroll
        for (int j = 0; j < 8; ++j) {
            float m0 = M0[j], m1 = M1[j];
            float x0 = __shfl_xor(m0, 16, 32);   // M0 row j+8 -> lanes < 16
            float x1 = __shfl_xor(m1, 16, 32);   // M1 row j   -> lanes >= 16
            col[j]     = laneHi ? x1 : m0;
            col[j + 8] = laneHi ? m1 : x0;
        }

        // ---- Phase 2: one-sided Jacobi (Hestenes), xor parallel ordering.
        // Each 16-lane half handles one matrix; pairs (l, l^s) stay in-half.
        for (int sweep = 0; sweep < NSWEEPS; ++sweep) {
            for (int s = 1; s < 16; ++s) {
                float part[16];
                float alpha = 0.0f, beta = 0.0f, gamma = 0.0f;
                #pragma unroll
                for (int j = 0; j < 16; ++j) {
                    part[j] = __shfl_xor(col[j], s, 32);
                    alpha = fmaf(col[j],  col[j],  alpha);
                    beta  = fmaf(part[j], part[j], beta);
                    gamma = fmaf(col[j],  part[j], gamma);
                }
                const bool isLo = lane < (lane ^ s);
                float app = isLo ? alpha : beta;
                float aqq = isLo ? beta  : alpha;
                bool tiny = fabsf(gamma) < 1e-30f;
                float g    = tiny ? 1.0f : gamma;
                float zeta = (aqq - app) / (2.0f * g);
                float t    = copysignf(1.0f / (fabsf(zeta) +
                                     sqrtf(1.0f + zeta * zeta)), zeta);
                float cth  = 1.0f / sqrtf(1.0f + t * t);
                float sth  = cth * t;
                cth = tiny ? 1.0f : cth;
                sth = tiny ? 0.0f : sth;
                #pragma unroll
                for (int j = 0; j < 16; ++j) {
                    float u = col[j], v = part[j];
                    col[j] = isLo ? fmaf(cth, u, -sth * v)
                                  : fmaf(sth, v,  cth * u);
                }
            }
        }

        // ---- Singular value = column norm ----
        float s2 = 0.0f;
        #pragma unroll
        for (int j = 0; j < 16; ++j) s2 = fmaf(col[j], col[j], s2);
        float sv = sqrtf(s2);

        // ---- Bitonic sort descending across 16 lanes (per half) ----
        #pragma unroll
        for (int k = 2; k <= 16; k <<= 1) {
            #pragma unroll
            for (int j = k >> 1; j >= 1; j >>= 1) {
                float w  = __shfl_xor(sv, j, 32);
                bool up    = ((laneLo & k) == 0);
                bool lower = ((laneLo & j) == 0);
                float mn = fminf(sv, w), mx = fmaxf(sv, w);
                sv = (up == lower) ? mx : mn;
            }
        }

        // ---- Weighted dissimilarity + 16-lane butterfly reduction ----
        float term = relk * (1.0f - sv);
        term += __shfl_xor(term, 1, 32);
        term += __shfl_xor(term, 2, 32);
        term += __shfl_xor(term, 4, 32);
        term += __shfl_xor(term, 8, 32);

        if (laneLo == 0) {
            int p = p0 + pp * 2 + laneHi;
            out[(size_t)b * SZ_P + p] = term;
        }
    }
}

extern "C" void kernel_launch(void* const* d_in, const int* in_sizes, int n_in,
                              void* d_out, int out_size, void* d_ws, size_t ws_size,
                              hipStream_t stream) {
    const float* xs  = (const float*)d_in[0];   // [1024, 256, 16] f32
    const float* xp  = (const float*)d_in[1];   // [500, 256, 16]  f32
    const float* rel = (const float*)d_in[2];   // [1, 16]         f32
    float* out = (float*)d_out;                 // [1024, 500]     f32

    // 1024 b * 125 p-tiles = 128000 waves, 8 waves (256 thr) per block
    dim3 grid(16000), block(256);
    hipLaunchKernelGGL(proto_grassmann_kernel, grid, block, 0, stream,
                       xs, xp, rel, out);
}